// MultiFreqEncoder_8581344657662
// MI455X (gfx1250) — compile-verified
//
#include <hip/hip_runtime.h>
#include <cstdint>

// MultiFreqEncoder for MI455X (gfx1250).
// Memory-bound: 1.01 GB of output (43 us floor @ 23.3 TB/s). Feature tables
// total 768 KB (L2-resident); scales 0..6 (95.25 KB) are staged into LDS with
// CDNA5 async-to-LDS loads, transposed on the fly to channel-contiguous layout.

#define N_SCALES   10
#define N_STAGED   7            // scales 0..6 -> LDS (192*(2^7-1)*4 B = 95.25 KB)
#define BLOCK      256
#define OUT_COLS   483
#define LDS_FLOATS (192 * 127)  // 24384 floats

struct FeatPtrs { const float* p[N_SCALES]; };

__global__ __launch_bounds__(BLOCK)
void mfenc_kernel(const float* __restrict__ x,
                  float* __restrict__ out,
                  FeatPtrs fp, int npts)
{
    __shared__ __align__(16) float smem[LDS_FLOATS];
    const int tid = threadIdx.x;
    const uint32_t ldsBase = (uint32_t)(uintptr_t)(&smem[0]); // low 32 bits == LDS byte offset

    // ---- Stage scales 0..6 into LDS, transposing [b][c][h] -> [b][h][c] ----
    // Per-lane LDS + global addresses let the async copy do the permutation.
    #pragma unroll
    for (int s = 0; s < N_STAGED; ++s) {
        const int log2H = 2 + s;
        const int H = 1 << log2H;
        const int nElems = 48 * H;                       // 6*8*H floats
        const uint32_t scaleBaseF = 192u * ((1u << s) - 1u);
        const uint64_t g = (uint64_t)(uintptr_t)fp.p[s];
        for (int idx = tid; idx < nElems; idx += BLOCK) {
            const int h = idx & (H - 1);
            const int c = (idx >> log2H) & 7;
            const int b = idx >> (log2H + 3);
            const uint32_t ldsAddr = ldsBase +
                4u * (scaleBaseF + (uint32_t)(((((b << log2H) + h) << 3) + c)));
            const uint32_t goff = (uint32_t)(idx << 2);
            asm volatile("global_load_async_to_lds_b32 %0, %1, %2"
                         :: "v"(ldsAddr), "v"(goff), "s"(g)
                         : "memory");
        }
    }
    asm volatile("s_wait_asynccnt 0x0" ::: "memory");
    __syncthreads();

    const int n = blockIdx.x * BLOCK + tid;
    if (n >= npts) return;

    const float x0 = x[3 * n + 0];
    const float x1 = x[3 * n + 1];
    const float x2 = x[3 * n + 2];
    const float xd[3] = {x0, x1, x2};
    float* __restrict__ row = out + (size_t)n * OUT_COLS;
    row[0] = x0; row[1] = x1; row[2] = x2;

    const float HALF_PI = 1.57079632679489661923f;

    // ---------------- Scales 0..6: gather from LDS ----------------
    for (int s = 0; s < N_STAGED; ++s) {
        const int log2H = 2 + s;
        const int H = 1 << log2H;
        const float scale = (float)(1 << s);
        const float hm1 = (float)(H - 1);
        float lat[2][3];
        #pragma unroll
        for (int d = 0; d < 3; ++d) {
            const float t = xd[d] * scale;
            lat[0][d] = sinf(t);
            lat[1][d] = sinf(t + HALF_PI);   // matches reference: sin(t + 0.5*pi) in f32
        }
        #pragma unroll
        for (int p = 0; p < 2; ++p) {
            #pragma unroll
            for (int d = 0; d < 3; ++d) {
                const int b = p * 3 + d;
                const float coord = lat[p][d];
                const float iy  = (coord + 1.0f) * 0.5f * hm1;
                const float i0f = floorf(iy);
                const float w1  = iy - i0f;
                const float w0  = 1.0f - w1;
                int i0 = (int)i0f;
                i0 = i0 < 0 ? 0 : (i0 > H - 1 ? H - 1 : i0);
                int i1 = i0 + 1; if (i1 > H - 1) i1 = H - 1;

                const float* basep = smem + 192u * ((1u << s) - 1u) + ((b << log2H) << 3);
                const float4 a0 = *(const float4*)(basep + (i0 << 3));
                const float4 a1 = *(const float4*)(basep + (i0 << 3) + 4);
                const float4 b0 = *(const float4*)(basep + (i1 << 3));
                const float4 b1 = *(const float4*)(basep + (i1 << 3) + 4);
                const float v0[8] = {a0.x, a0.y, a0.z, a0.w, a1.x, a1.y, a1.z, a1.w};
                const float v1[8] = {b0.x, b0.y, b0.z, b0.w, b1.x, b1.y, b1.z, b1.w};

                float* orow = row + 3 + s * 6 + p * 3 + d;
                #pragma unroll
                for (int c = 0; c < 8; ++c)
                    orow[c * 60] = (w0 * v0[c] + w1 * v1[c]) + coord;
            }
        }
    }

    // ---------------- Scales 7..9: gather from global (L2-resident) ----------------
    #pragma unroll
    for (int si = 0; si < 3; ++si) {
        const int s = N_STAGED + si;
        const int H = 4 << s;
        const float scale = (float)(1 << s);
        const float hm1 = (float)(H - 1);
        float lat[2][3];
        #pragma unroll
        for (int d = 0; d < 3; ++d) {
            const float t = xd[d] * scale;
            lat[0][d] = sinf(t);
            lat[1][d] = sinf(t + HALF_PI);
        }
        const float* __restrict__ ft = fp.p[s];
        #pragma unroll
        for (int p = 0; p < 2; ++p) {
            #pragma unroll
            for (int d = 0; d < 3; ++d) {
                const int b = p * 3 + d;
                const float coord = lat[p][d];
                const float iy  = (coord + 1.0f) * 0.5f * hm1;
                const float i0f = floorf(iy);
                const float w1  = iy - i0f;
                const float w0  = 1.0f - w1;
                int i0 = (int)i0f;
                i0 = i0 < 0 ? 0 : (i0 > H - 1 ? H - 1 : i0);
                int i1 = i0 + 1; if (i1 > H - 1) i1 = H - 1;

                const float* basep = ft + b * 8 * H;   // layout [b][c][h]
                float* orow = row + 3 + s * 6 + p * 3 + d;
                #pragma unroll
                for (int c = 0; c < 8; ++c) {
                    const float v0 = basep[c * H + i0];
                    const float v1 = basep[c * H + i1];
                    orow[c * 60] = (w0 * v0 + w1 * v1) + coord;
                }
            }
        }
    }
}

extern "C" void kernel_launch(void* const* d_in, const int* in_sizes, int n_in,
                              void* d_out, int out_size, void* d_ws, size_t ws_size,
                              hipStream_t stream)
{
    (void)d_ws; (void)ws_size; (void)out_size;
    const float* x = (const float*)d_in[0];

    FeatPtrs fp;
    if (n_in >= 1 + N_SCALES) {
        // features tuple flattened into separate inputs d_in[1..10]
        for (int i = 0; i < N_SCALES; ++i) fp.p[i] = (const float*)d_in[1 + i];
    } else {
        // features concatenated flat in d_in[1]; offset before scale i = 192*(2^i - 1) elems
        const float* base = (const float*)d_in[1];
        for (int i = 0; i < N_SCALES; ++i) fp.p[i] = base + 192u * ((1u << i) - 1u);
    }

    const int npts = in_sizes[0] / 3;
    dim3 grid((npts + BLOCK - 1) / BLOCK), block(BLOCK);
    hipLaunchKernelGGL(mfenc_kernel, grid, block, 0, stream,
                       x, (float*)d_out, fp, npts);
}